// PointAttention_83425444757522
// MI455X (gfx1250) — compile-verified
//
#include <hip/hip_runtime.h>

#define Bsz   8
#define Lseq  1024
#define LOseq 1024
#define Hdim  768
#define Vdim  4096
#define OUTW  (Vdim + Lseq)   // 5120

typedef float v2f __attribute__((ext_vector_type(2)));
typedef float v8f __attribute__((ext_vector_type(8)));

// ---------------------------------------------------------------------------
// Kernel 1: per-batch run segmentation of sorted tags.
// One 1024-thread block per batch. Hillis-Steele inclusive scan in LDS.
// ---------------------------------------------------------------------------
__global__ void __launch_bounds__(Lseq)
seg_kernel(const int* __restrict__ tags,
           int* __restrict__ seg_start,
           int* __restrict__ counts,
           int* __restrict__ emitted) {
    const int b = blockIdx.x;
    const int i = threadIdx.x;                 // 0..L-1
    __shared__ int scan[Lseq];
    __shared__ int cnt[Lseq];

    const int* t    = tags + b * Lseq;
    const int tag   = t[i];
    const int valid = (tag != -1);
    const int prev  = (i == 0) ? -2 : t[i - 1];
    const int start = valid && ((i == 0) || (tag != prev) || (prev == -1));

    scan[i] = start;
    cnt[i]  = 0;
    __syncthreads();
    for (int off = 1; off < Lseq; off <<= 1) {
        int v   = scan[i];
        int add = (i >= off) ? scan[i - off] : 0;
        __syncthreads();
        scan[i] = v + add;
        __syncthreads();
    }
    const int seg = scan[i] - 1;               // run index of this position
    if (valid) atomicAdd(&cnt[seg], 1);
    if (start) seg_start[b * Lseq + seg] = i;
    __syncthreads();
    counts[b * Lseq + i] = cnt[i];
    if (i == Lseq - 1) {
        // trailing open run (still running at last position) is dropped
        emitted[b] = scan[Lseq - 1] - (valid ? 1 : 0);
    }
}

// ---------------------------------------------------------------------------
// Kernel 2: attn[b,r,:] = mean of hidden1 rows in segment r (zero if r>=emitted)
// One 256-thread block per (row, batch).
// ---------------------------------------------------------------------------
__global__ void __launch_bounds__(256)
mean_kernel(const float* __restrict__ hidden1,
            const int* __restrict__ seg_start,
            const int* __restrict__ counts,
            const int* __restrict__ emitted,
            float* __restrict__ attn) {
    const int b = blockIdx.y;
    const int r = blockIdx.x;
    const int h = threadIdx.x;
    float* dst = attn + ((size_t)b * Lseq + r) * Hdim;

    const int em = emitted[b];
    if (r >= em) {
        for (int j = h; j < Hdim; j += 256) dst[j] = 0.0f;
        return;
    }
    const int s   = seg_start[b * Lseq + r];
    const int len = counts[b * Lseq + r];
    const float inv = 1.0f / (float)(len > 0 ? len : 1);
    const float* src = hidden1 + ((size_t)b * Lseq + s) * Hdim;
    for (int j = h; j < Hdim; j += 256) {
        float acc = 0.0f;
        for (int k = 0; k < len; ++k) acc += src[(size_t)k * Hdim + j];
        dst[j] = acc * inv;
    }
}

// ---------------------------------------------------------------------------
// Kernel 3: attn1[b] = hidden2[b] (LOxH) @ attn[b]^T (HxL) + bias, written to
// out[b, o, V + l].  64x64 tile per block, LDS-staged, v_wmma_f32_16x16x4_f32.
// 8 waves; wave w owns subtile row (w&3) and subtile cols {2*(w>>2), +1}.
// ---------------------------------------------------------------------------
__global__ void __launch_bounds__(256)
gemm_bias_kernel(const float* __restrict__ hidden2,
                 const float* __restrict__ attn,
                 const int* __restrict__ emitted,
                 float* __restrict__ out) {
    const int b  = blockIdx.z;
    const int o0 = blockIdx.y * 64;
    const int l0 = blockIdx.x * 64;

    // stride 68 floats: keeps 16B alignment for float4 stores and spreads the
    // wmma fragment ds_load_b64 reads across all 64 LDS banks (row*4+col mod 64)
    __shared__ __align__(16) float As[64][68];
    __shared__ __align__(16) float Bs[64][68];

    const int tid   = threadIdx.x;
    const int wave  = tid >> 5;
    const int lane  = tid & 31;
    const int r     = wave & 3;          // subtile row 0..3
    const int c0    = (wave >> 2) << 1;  // subtile col base: 0 or 2
    const int l15   = lane & 15;
    const int khalf = lane >> 4;         // ISA: lanes 16-31 carry K upper half

    v8f acc0 = {0.f, 0.f, 0.f, 0.f, 0.f, 0.f, 0.f, 0.f};
    v8f acc1 = {0.f, 0.f, 0.f, 0.f, 0.f, 0.f, 0.f, 0.f};

    const float* Aglob = hidden2 + ((size_t)b * LOseq + o0) * Hdim;
    const float* Bglob = attn    + ((size_t)b * Lseq  + l0) * Hdim;

    for (int k0 = 0; k0 < Hdim; k0 += 64) {
        // stage 64x64 chunks of A (hidden2) and B (attn) with float4 loads
        for (int idx = tid; idx < 64 * 16; idx += 256) {
            const int row = idx >> 4;
            const int c4  = (idx & 15) << 2;
            float4 va = *(const float4*)(Aglob + (size_t)row * Hdim + k0 + c4);
            float4 vb = *(const float4*)(Bglob + (size_t)row * Hdim + k0 + c4);
            *(float4*)&As[row][c4] = va;
            *(float4*)&Bs[row][c4] = vb;
        }
        __syncthreads();

#pragma unroll
        for (int kk = 0; kk < 64; kk += 4) {
            const int ka = kk + (khalf << 1);   // lanes 0-15: K=kk,kk+1 ; 16-31: K=kk+2,kk+3
            v2f afrag;
            afrag.x = As[(r << 4) + l15][ka];
            afrag.y = As[(r << 4) + l15][ka + 1];
            v2f bf0;
            bf0.x = Bs[(c0 << 4) + l15][ka];
            bf0.y = Bs[(c0 << 4) + l15][ka + 1];
            v2f bf1;
            bf1.x = Bs[((c0 + 1) << 4) + l15][ka];
            bf1.y = Bs[((c0 + 1) << 4) + l15][ka + 1];
            acc0 = __builtin_amdgcn_wmma_f32_16x16x4_f32(
                false, afrag, false, bf0, (short)0, acc0, false, false);
            acc1 = __builtin_amdgcn_wmma_f32_16x16x4_f32(
                false, afrag, false, bf1, (short)0, acc1, false, false);
        }
        __syncthreads();
    }

    // epilogue: D element (M = v + 8*khalf, N = l15); fuse bias, store to out
    const int em    = emitted[b];
    const int lcol0 = l0 + (c0 << 4) + l15;
    const int lcol1 = lcol0 + 16;
    const float bias0 = (lcol0 < em) ? 0.0f : -100.0f;
    const float bias1 = (lcol1 < em) ? 0.0f : -100.0f;
#pragma unroll
    for (int v = 0; v < 8; ++v) {
        const int orow = o0 + (r << 4) + v + (khalf << 3);
        float* dst = out + ((size_t)b * LOseq + orow) * OUTW + Vdim;
        dst[lcol0] = acc0[v] + bias0;
        dst[lcol1] = acc1[v] + bias1;
    }
}

// ---------------------------------------------------------------------------
// Kernel 4: out[b, o, 0:V] = lm_logits[b, o, :]   (float4 copy)
// ---------------------------------------------------------------------------
__global__ void __launch_bounds__(256)
copy_logits_kernel(const float4* __restrict__ lm, float* __restrict__ out) {
    const size_t total = (size_t)Bsz * LOseq * (Vdim / 4);
    size_t i = (size_t)blockIdx.x * blockDim.x + threadIdx.x;
    if (i >= total) return;
    float4 v = lm[i];
    const size_t row = i / (Vdim / 4);     // b*LO + o
    const size_t c4  = i % (Vdim / 4);
    *(float4*)(out + row * OUTW + (c4 << 2)) = v;
}

// ---------------------------------------------------------------------------
extern "C" void kernel_launch(void* const* d_in, const int* in_sizes, int n_in,
                              void* d_out, int out_size, void* d_ws, size_t ws_size,
                              hipStream_t stream) {
    const float* hidden1 = (const float*)d_in[0];
    const float* hidden2 = (const float*)d_in[1];
    const float* lm      = (const float*)d_in[2];
    const int*   tags    = (const int*)d_in[3];
    float* out = (float*)d_out;

    // workspace layout
    float* attn     = (float*)d_ws;                                   // B*L*H f32
    int*   segstart = (int*)(attn + (size_t)Bsz * Lseq * Hdim);       // B*L
    int*   counts   = segstart + Bsz * Lseq;                          // B*L
    int*   emitted  = counts + Bsz * Lseq;                            // B

    seg_kernel<<<Bsz, Lseq, 0, stream>>>(tags, segstart, counts, emitted);
    mean_kernel<<<dim3(Lseq, Bsz), 256, 0, stream>>>(hidden1, segstart, counts,
                                                     emitted, attn);
    gemm_bias_kernel<<<dim3(Lseq / 64, LOseq / 64, Bsz), 256, 0, stream>>>(
        hidden2, attn, emitted, out);

    const size_t nf4 = (size_t)Bsz * LOseq * (Vdim / 4);
    copy_logits_kernel<<<(unsigned)((nf4 + 255) / 256), 256, 0, stream>>>(
        (const float4*)lm, out);
}